// GINModel_77395310674454
// MI455X (gfx1250) — compile-verified
//
#include <hip/hip_runtime.h>

#define NNODES 50000
#define NEDGES 800000
#define FD     128      // IN == HID == 128
#define NOUT   64
#define NGRAPH 256
#define BN_EPS 1e-5f

typedef __attribute__((ext_vector_type(2))) float v2f;
typedef __attribute__((ext_vector_type(8))) float v8f;

// ---------------------------------------------------------------- utilities
__global__ void fill_zero(float* __restrict__ p, int n) {
  int i = blockIdx.x * blockDim.x + threadIdx.x;
  int stride = gridDim.x * blockDim.x;
  for (; i < n; i += stride) p[i] = 0.f;
}

__global__ void copy_vec(const float4* __restrict__ s, float4* __restrict__ d, int n4) {
  int i = blockIdx.x * blockDim.x + threadIdx.x;
  int stride = gridDim.x * blockDim.x;
  for (; i < n4; i += stride) d[i] = s[i];
}

// ------------------------------------------------- edge scatter: z[dst] += h[src]
// block = 256 threads = 2 edges x 128 features (coalesced 512B row per edge)
__global__ void edge_scatter(float* __restrict__ z, const float* __restrict__ h,
                             const int* __restrict__ src, const int* __restrict__ dst) {
  int e = blockIdx.x * 2 + (threadIdx.x >> 7);
  int f = threadIdx.x & 127;
  if (e >= NEDGES) return;
  int s = __ldg(&src[e]);
  int d = __ldg(&dst[e]);
  atomicAdd(&z[(size_t)d * FD + f], h[(size_t)s * FD + f]);
}

// -------------------------------------------------------------- BN finalize
// scale = g * rsqrt(var+eps); shift = b - mean*scale   (128 threads, 1 block)
__global__ void finalize_bn(const float* __restrict__ sum, const float* __restrict__ sq,
                            const float* __restrict__ g, const float* __restrict__ b,
                            float* __restrict__ scale, float* __restrict__ shift, float invN) {
  int i = threadIdx.x;
  float m  = sum[i] * invN;
  float v  = sq[i] * invN - m * m;
  float rs = rsqrtf(v + BN_EPS);
  float sc = g[i] * rs;
  scale[i] = sc;
  shift[i] = b[i] - m * sc;
}

// ---------------------- pointwise: out = relu(in*scale+shift), accumulate stats
__global__ void pw_bn_relu_stats(float* __restrict__ out, const float* __restrict__ in,
                                 const float* __restrict__ scale, const float* __restrict__ shift,
                                 float* __restrict__ sum, float* __restrict__ sq, int M) {
  int col = threadIdx.x;                 // 0..127
  float sc = scale[col], sh = shift[col];
  float s = 0.f, q = 0.f;
  for (int r = blockIdx.x; r < M; r += gridDim.x) {
    float v = fmaxf(in[(size_t)r * FD + col] * sc + sh, 0.f);
    out[(size_t)r * FD + col] = v;
    s += v; q += v * v;
  }
  atomicAdd(&sum[col], s);
  atomicAdd(&sq[col], q);
}

// ------------- pointwise: h = relu(in*scale+shift) + sorted-segment pooling
// contiguous row chunk per block; flush one atomic per graph boundary per column
__global__ void pw_bn_relu_pool(float* __restrict__ out, float* __restrict__ pooled,
                                const float* __restrict__ in,
                                const float* __restrict__ scale, const float* __restrict__ shift,
                                const int* __restrict__ batch, int M, int rowsPerBlock) {
  int col = threadIdx.x;
  int r0 = blockIdx.x * rowsPerBlock;
  int r1 = min(M, r0 + rowsPerBlock);
  if (r0 >= r1) return;
  float sc = scale[col], sh = shift[col];
  int cur = batch[r0];
  float acc = 0.f;
  for (int r = r0; r < r1; ++r) {
    int b = batch[r];
    if (b != cur) { atomicAdd(&pooled[(size_t)cur * FD + col], acc); acc = 0.f; cur = b; }
    float v = fmaxf(in[(size_t)r * FD + col] * sc + sh, 0.f);
    out[(size_t)r * FD + col] = v;
    acc += v;
  }
  atomicAdd(&pooled[(size_t)cur * FD + col], acc);
}

// ------------------------------- sorted-segment pooling only (layer-0 features)
__global__ void pool_rows(float* __restrict__ pooled, const float* __restrict__ x,
                          const int* __restrict__ batch, int M, int rowsPerBlock) {
  int col = threadIdx.x;
  int r0 = blockIdx.x * rowsPerBlock;
  int r1 = min(M, r0 + rowsPerBlock);
  if (r0 >= r1) return;
  int cur = batch[r0];
  float acc = 0.f;
  for (int r = r0; r < r1; ++r) {
    int b = batch[r];
    if (b != cur) { atomicAdd(&pooled[(size_t)cur * FD + col], acc); acc = 0.f; cur = b; }
    acc += x[(size_t)r * FD + col];
  }
  atomicAdd(&pooled[(size_t)cur * FD + col], acc);
}

// ---------------------------------------------------------------- FP32 WMMA GEMM
// Y[M x N] = (INBN ? relu(X*iscale+ishift) : X)[M x K] @ W[K x N] + bias (+Y if ACC)
// One wave computes a 16-row strip across all NT 16-col tiles (A fragment reuse).
// STATS: accumulate per-column sum / sumsq of Y into sum/sq (for BatchNorm).
// V_WMMA_F32_16X16X4_F32 fragment layout (ISA 7.12.2):
//   A (16x4): lanes 0-15 -> {K0,K1}, lanes 16-31 -> {K2,K3}, M = lane&15
//   C/D     : VGPR r -> M = r + 8*(lane>>4), N = lane&15
template <int NT, bool INBN, bool STATS, bool ACC>
__global__ void gemm_wmma(float* __restrict__ Y, const float* __restrict__ X,
                          const float* __restrict__ W, const float* __restrict__ bias,
                          const float* __restrict__ iscale, const float* __restrict__ ishift,
                          float* __restrict__ sum, float* __restrict__ sq,
                          int M, int K) {
  constexpr int N = NT * 16;
  const int wave = threadIdx.x >> 5;
  const int lane = threadIdx.x & 31;
  const int tile = blockIdx.x * (blockDim.x >> 5) + wave;   // M-tile index (wave-uniform)
  const int mTiles = M >> 4;
  if (tile >= mTiles) return;                               // whole wave exits: EXEC stays full
  const int m0   = tile << 4;
  const int half = lane >> 4;
  const int l15  = lane & 15;

  v8f zero = {};
  v8f c[NT];
#pragma unroll
  for (int t = 0; t < NT; ++t) c[t] = zero;

  const float* xrow = X + (size_t)(m0 + l15) * K;
  for (int k = 0; k < K; k += 4) {
    int ka = k + 2 * half;
    float a0 = xrow[ka];
    float a1 = xrow[ka + 1];
    if (INBN) {
      a0 = fmaxf(a0 * iscale[ka] + ishift[ka], 0.f);
      a1 = fmaxf(a1 * iscale[ka + 1] + ishift[ka + 1], 0.f);
    }
    v2f a; a.x = a0; a.y = a1;
#pragma unroll
    for (int t = 0; t < NT; ++t) {
      int n = t * 16 + l15;
      v2f b; b.x = W[(size_t)ka * N + n]; b.y = W[(size_t)(ka + 1) * N + n];
      c[t] = __builtin_amdgcn_wmma_f32_16x16x4_f32(false, a, false, b,
                                                   (short)0, c[t], false, false);
    }
  }

#pragma unroll
  for (int t = 0; t < NT; ++t) {
    int n = t * 16 + l15;
    float bv = bias[n];
    float s = 0.f, q = 0.f;
#pragma unroll
    for (int r = 0; r < 8; ++r) {
      int m = m0 + r + 8 * half;
      float v = c[t][r] + bv;
      if (ACC) v += Y[(size_t)m * N + n];
      Y[(size_t)m * N + n] = v;
      if (STATS) { s += v; q += v * v; }
    }
    if (STATS) { atomicAdd(&sum[n], s); atomicAdd(&sq[n], q); }
  }
}

// ---------------------------------------------------------------------------
// d_in flattening (setup_inputs insertion order):
//   0: n_feat | 1+10*l+j: gin[l]{W1,b1,g1,be1,W2,b2,ga,ba,go,bo} | 41+2*i+j: pred[i]{W,b}
//   51: edge_src | 52: edge_dst | 53: batch
// d_out: score(256x64) ++ pooled[layer1..4](each 256x128)
extern "C" void kernel_launch(void* const* d_in, const int* in_sizes, int n_in,
                              void* d_out, int out_size, void* d_ws, size_t ws_size,
                              hipStream_t stream) {
  (void)in_sizes; (void)n_in; (void)out_size; (void)ws_size;

  const float* n_feat   = (const float*)d_in[0];
  const int*   edge_src = (const int*)d_in[51];
  const int*   edge_dst = (const int*)d_in[52];
  const int*   batch    = (const int*)d_in[53];
  auto gin  = [&](int l, int j) { return (const float*)d_in[1 + 10 * l + j]; };
  auto pred = [&](int i, int j) { return (const float*)d_in[41 + 2 * i + j]; };

  // workspace carve-up (floats)
  float* B0      = (float*)d_ws;                 // h_new / h_in (25.6 MB)
  float* B1      = B0 + (size_t)NNODES * FD;     // z, then y2
  float* B2      = B1 + (size_t)NNODES * FD;     // y1, then a
  float* sum     = B2 + (size_t)NNODES * FD;     // 128
  float* sq      = sum + FD;                     // 128
  float* scale   = sq + FD;                      // 128
  float* shift   = scale + FD;                   // 128
  float* pooled0 = shift + FD;                   // 256*128

  float* score = (float*)d_out;
  const int SCORE_SZ  = NGRAPH * NOUT;           // 16384
  const int POOLED_SZ = NGRAPH * FD;             // 32768

  // zero outputs + pooled0 (harness poisons once; we rebuild every call)
  fill_zero<<<256, 256, 0, stream>>>(score, SCORE_SZ + 4 * POOLED_SZ);
  fill_zero<<<128, 256, 0, stream>>>(pooled0, POOLED_SZ);

  const int chunk = (NNODES + 1023) / 1024;      // rows per pooling block
  pool_rows<<<1024, FD, 0, stream>>>(pooled0, n_feat, batch, NNODES, chunk);

  const int mTiles     = NNODES / 16;            // 3125
  const int gemmBlocks = (mTiles + 7) / 8;       // 8 waves per 256-thread block
  const float invN     = 1.f / (float)NNODES;

  const float* h = n_feat;
  for (int l = 0; l < 4; ++l) {
    float* z = B1, *y1 = B2, *y2 = B1, *a = B2, *hn = B0;

    // z = h + scatter_add(h[src] -> dst)
    copy_vec<<<1024, 256, 0, stream>>>((const float4*)h, (float4*)z, NNODES * FD / 4);
    edge_scatter<<<NEDGES / 2, 256, 0, stream>>>(z, h, edge_src, edge_dst);

    // y1 = z @ W1 + b1   (+ column stats)
    fill_zero<<<1, 256, 0, stream>>>(sum, 2 * FD);
    gemm_wmma<8, false, true, false><<<gemmBlocks, 256, 0, stream>>>(
        y1, z, gin(l, 0), gin(l, 1), nullptr, nullptr, sum, sq, NNODES, FD);
    finalize_bn<<<1, FD, 0, stream>>>(sum, sq, gin(l, 2), gin(l, 3), scale, shift, invN);

    // y2 = relu(bn(y1)) @ W2 + b2   (BN+ReLU fused into A-fragment load; + stats)
    fill_zero<<<1, 256, 0, stream>>>(sum, 2 * FD);
    gemm_wmma<8, true, true, false><<<gemmBlocks, 256, 0, stream>>>(
        y2, y1, gin(l, 4), gin(l, 5), scale, shift, sum, sq, NNODES, FD);
    finalize_bn<<<1, FD, 0, stream>>>(sum, sq, gin(l, 6), gin(l, 7), scale, shift, invN);

    // a = relu(bn(y2))   (+ stats of a)
    fill_zero<<<1, 256, 0, stream>>>(sum, 2 * FD);
    pw_bn_relu_stats<<<1024, FD, 0, stream>>>(a, y2, scale, shift, sum, sq, NNODES);
    finalize_bn<<<1, FD, 0, stream>>>(sum, sq, gin(l, 8), gin(l, 9), scale, shift, invN);

    // h = relu(bn(a)) + segment-sum pooling into d_out
    float* pooled_l = score + SCORE_SZ + l * POOLED_SZ;
    pw_bn_relu_pool<<<1024, FD, 0, stream>>>(hn, pooled_l, a, scale, shift, batch,
                                             NNODES, chunk);
    h = hn;
  }

  // score = sum_i pooled_i @ predW_i + predb_i   (accumulating WMMA GEMMs)
  gemm_wmma<4, false, false, true><<<2, 256, 0, stream>>>(
      score, pooled0, pred(0, 0), pred(0, 1), nullptr, nullptr, nullptr, nullptr,
      NGRAPH, FD);
  for (int i = 1; i < 5; ++i) {
    const float* pl = score + SCORE_SZ + (i - 1) * POOLED_SZ;
    gemm_wmma<4, false, false, true><<<2, 256, 0, stream>>>(
        score, pl, pred(i, 0), pred(i, 1), nullptr, nullptr, nullptr, nullptr,
        NGRAPH, FD);
  }
}